// PoincareEncoder_13615046328718
// MI455X (gfx1250) — compile-verified
//
#include <hip/hip_runtime.h>
#include <math.h>

// ---------------- problem constants ----------------
#define BS_    4
#define T_     64
#define NN_    128
#define NF_    16
#define HID_   64
#define HEADS_ 4
#define RH_    256
#define LD_    64
#define EPG_   1024
#define NG_    (BS_ * T_)          // 256 graphs
#define NNODE_ (NG_ * NN_)         // 32768 nodes
#define NEDGE_ (NG_ * EPG_)        // 262144 batched edges
#define ETOT_  (NEDGE_ + NNODE_)   // + self loops = 294912

#define GEMM_WAVES   8             // waves per GEMM block
#define GEMM_MT      2             // 16-row M tiles per wave
#define GEMM_ROWS    (16 * GEMM_WAVES * GEMM_MT)   // 256 rows per block
#define KMAX_        256           // max K (layers 1/2)

typedef __attribute__((ext_vector_type(2))) float v2f;
typedef __attribute__((ext_vector_type(8))) float v8f;

// ---------------- small helpers ----------------
__device__ __forceinline__ unsigned fenc(float f) {
    int i = __float_as_int(f);
    return (unsigned)i ^ ((unsigned)(i >> 31) | 0x80000000u);
}
__device__ __forceinline__ float fdec(unsigned k) {
    int i = (k & 0x80000000u) ? (int)(k ^ 0x80000000u) : (int)~k;
    return __int_as_float(i);
}
__device__ __forceinline__ void edge_sd(const int* __restrict__ ei, int e, int& s, int& d) {
    if (e < NEDGE_) {
        int g  = e >> 10;          // / EPG_
        int be = e & (EPG_ - 1);
        s = ei[be]        + g * NN_;
        d = ei[EPG_ + be] + g * NN_;
    } else {
        s = d = e - NEDGE_;        // self loop
    }
}
__device__ __forceinline__ float sigmoidf_(float x) { return 1.0f / (1.0f + expf(-x)); }

// ---------------- fills ----------------
__global__ void fill_f32(float* p, float v, int n) {
    int i = blockIdx.x * blockDim.x + threadIdx.x;
    if (i < n) p[i] = v;
}
__global__ void fill_u32(unsigned* p, unsigned v, int n) {
    int i = blockIdx.x * blockDim.x + threadIdx.x;
    if (i < n) p[i] = v;
}

// ---------------- fp32 WMMA GEMM: C[M x Nc] = A[M x K] * W[K x Nc] ----------------
// Block = 8 waves, each wave owns two 16x16 output tiles in the same 16-wide N
// column (ntile = blockIdx.x). The W column block [K x 16] is staged once into
// LDS, pair-swizzled so each lane's B fragment is one aligned ds_load_b64:
//   lds_b[(k>>1)*32 + col*2 + (k&1)] = W[k][ntile*16 + col]
// Inner K-step per wave: 2x global_load_b64 (A) + 1x ds_load_b64 (B) + 2x WMMA.
__global__ __launch_bounds__(32 * GEMM_WAVES) void wmma_matmul(
        const float* __restrict__ A, const float* __restrict__ W,
        float* __restrict__ C, int K, int Nc) {
    __shared__ float lds_b[(KMAX_ / 2) * 32];   // 16 KB

    const int tid   = threadIdx.x;
    const int ntile = blockIdx.x;
    const int cbase = ntile * 16;

    // ---- stage W[:,cbase:cbase+16] into LDS (pair-swizzled) ----
    for (int idx = tid; idx < K * 16; idx += 32 * GEMM_WAVES) {
        int k = idx >> 4, col = idx & 15;
        lds_b[(k >> 1) * 32 + col * 2 + (k & 1)] = W[k * Nc + cbase + col];
    }
    __syncthreads();

    const int wave = tid >> 5;
    const int lane = tid & 31;
    const int half = lane >> 4;                  // 0: K pair 0-1, 1: K pair 2-3
    const int l    = lane & 15;

    const int row0 = blockIdx.y * GEMM_ROWS + wave * (16 * GEMM_MT) + l;  // tile 0
    const int row1 = row0 + 16;                                           // tile 1
    const float* a0p = A + (size_t)row0 * K;
    const float* a1p = A + (size_t)row1 * K;

    v8f acc0 = {}, acc1 = {};
    for (int k = 0; k < K; k += 4) {
        const int ka = k + half * 2;
        // B fragment from LDS: one aligned 8-byte read
        v2f b = *(const v2f*)&lds_b[(ka >> 1) * 32 + l * 2];
        // A fragments: contiguous pairs -> global_load_b64
        v2f a0 = *(const v2f*)&a0p[ka];
        v2f a1 = *(const v2f*)&a1p[ka];
        __builtin_prefetch(&a0p[ka + 4], 0, 1);
        acc0 = __builtin_amdgcn_wmma_f32_16x16x4_f32(false, a0, false, b,
                                                     (short)0, acc0, false, false);
        acc1 = __builtin_amdgcn_wmma_f32_16x16x4_f32(false, a1, false, b,
                                                     (short)0, acc1, false, false);
    }

    // C/D layout: VGPR r -> M = r (lanes 0-15) / M = 8+r (lanes 16-31), N = lane&15
    const int crow0 = blockIdx.y * GEMM_ROWS + wave * (16 * GEMM_MT) + half * 8;
    const int ccol  = cbase + l;
#pragma unroll
    for (int r = 0; r < 8; ++r) {
        C[(size_t)(crow0 + r) * Nc + ccol]      = acc0[r];
        C[(size_t)(crow0 + 16 + r) * Nc + ccol] = acc1[r];
    }
}

// ---------------- GAT attention pieces ----------------
// al_s[n,h] = <h[n,h,:], a_src[h,:]> ; al_d likewise
__global__ void attn_logits(const float* __restrict__ h,
                            const float* __restrict__ asrc,
                            const float* __restrict__ adst,
                            float* __restrict__ al_s, float* __restrict__ al_d, int H) {
    int i = blockIdx.x * blockDim.x + threadIdx.x;
    if (i >= NNODE_ * H) return;
    int n = i / H, hh = i % H;
    const float* hv = h + (n * H + hh) * HID_;
    const float* as = asrc + hh * HID_;
    const float* ad = adst + hh * HID_;
    float s = 0.f, d = 0.f;
#pragma unroll 8
    for (int k = 0; k < HID_; ++k) { s += hv[k] * as[k]; d += hv[k] * ad[k]; }
    al_s[i] = s; al_d[i] = d;
}

__global__ void edge_max(const int* __restrict__ ei, const float* __restrict__ al_s,
                         const float* __restrict__ al_d, unsigned* __restrict__ mkey, int H) {
    int e = blockIdx.x * blockDim.x + threadIdx.x;
    if (e >= ETOT_) return;
    int s, d; edge_sd(ei, e, s, d);
    for (int hh = 0; hh < H; ++hh) {
        float v = al_s[s * H + hh] + al_d[d * H + hh];
        v = (v > 0.f) ? v : 0.2f * v;                  // leaky_relu(., 0.2)
        atomicMax(&mkey[d * H + hh], fenc(v));
    }
}

__global__ void edge_den(const int* __restrict__ ei, const float* __restrict__ al_s,
                         const float* __restrict__ al_d, const unsigned* __restrict__ mkey,
                         float* __restrict__ den, int H) {
    int e = blockIdx.x * blockDim.x + threadIdx.x;
    if (e >= ETOT_) return;
    int s, d; edge_sd(ei, e, s, d);
    for (int hh = 0; hh < H; ++hh) {
        float v = al_s[s * H + hh] + al_d[d * H + hh];
        v = (v > 0.f) ? v : 0.2f * v;
        float ex = expf(v - fdec(mkey[d * H + hh]));
        atomicAdd(&den[d * H + hh], ex);
    }
}

__global__ void edge_agg(const int* __restrict__ ei, const float* __restrict__ h,
                         const float* __restrict__ al_s, const float* __restrict__ al_d,
                         const unsigned* __restrict__ mkey, const float* __restrict__ den,
                         float* __restrict__ agg, int H) {
    int i = blockIdx.x * blockDim.x + threadIdx.x;
    if (i >= ETOT_ * H) return;
    int e = i / H, hh = i % H;
    int s, d; edge_sd(ei, e, s, d);
    float v = al_s[s * H + hh] + al_d[d * H + hh];
    v = (v > 0.f) ? v : 0.2f * v;
    float ex = expf(v - fdec(mkey[d * H + hh]));
    float w  = ex / (den[d * H + hh] + 1e-16f);
    const float* hv = h  + (s * H + hh) * HID_;
    float* av       = agg + (d * H + hh) * HID_;
#pragma unroll 4
    for (int f = 0; f < HID_; ++f)
        atomicAdd(&av[f], hv[f] * w);
}

__global__ void bias_relu(float* __restrict__ a, const float* __restrict__ b, int C, int n) {
    int i = blockIdx.x * blockDim.x + threadIdx.x;
    if (i >= n) return;
    float v = a[i] + b[i % C];
    a[i] = (v > 0.f) ? v : 0.f;
}

// ---------------- graph pooling: embs[g,d] = sum_n h2[g*NN+n, d] ----------------
__global__ void pool_kernel(const float* __restrict__ h2, float* __restrict__ embs) {
    int g = blockIdx.x, d = threadIdx.x;     // 64 threads
    float s = 0.f;
    for (int n = 0; n < NN_; ++n) s += h2[(g * NN_ + n) * HID_ + d];
    embs[g * HID_ + d] = s;
}

// ---------------- bidirectional LSTM: one workgroup per direction ----------------
__global__ __launch_bounds__(1024) void lstm_kernel(
    const float* __restrict__ embs,       // [NG][HID], g = b*T + t
    const float* __restrict__ WihF, const float* __restrict__ WhhF,
    const float* __restrict__ bihF, const float* __restrict__ bhhF,
    const float* __restrict__ WihB, const float* __restrict__ WhhB,
    const float* __restrict__ bihB, const float* __restrict__ bhhB,
    float* __restrict__ feat)             // [BS][2*RH]
{
    __shared__ float hs[BS_][RH_];
    __shared__ float cs[BS_][RH_];
    __shared__ float xs[BS_][HID_];
    __shared__ float gs[BS_][4 * RH_];

    const int dir = blockIdx.x;
    const float* Wih = dir ? WihB : WihF;
    const float* Whh = dir ? WhhB : WhhF;
    const float* bih = dir ? bihB : bihF;
    const float* bhh = dir ? bhhB : bhhF;

    const int tid = threadIdx.x;          // 0..1023 == gate row j
    for (int i = tid; i < BS_ * RH_; i += 1024) {
        hs[i / RH_][i % RH_] = 0.f;
        cs[i / RH_][i % RH_] = 0.f;
    }
    __syncthreads();

    const float bsum = bih[tid] + bhh[tid];
    const float* wi = Wih + tid * HID_;
    const float* wh = Whh + tid * RH_;

    for (int step = 0; step < T_; ++step) {
        const int t = dir ? (T_ - 1 - step) : step;
        if (tid < BS_ * HID_) {
            int b = tid >> 6, d = tid & 63;
            xs[b][d] = embs[(b * T_ + t) * HID_ + d];
        }
        __syncthreads();

        float acc0 = bsum, acc1 = bsum, acc2 = bsum, acc3 = bsum;
        for (int k = 0; k < HID_; ++k) {
            float w = wi[k];
            acc0 += xs[0][k] * w; acc1 += xs[1][k] * w;
            acc2 += xs[2][k] * w; acc3 += xs[3][k] * w;
        }
        for (int k = 0; k < RH_; ++k) {
            float w = wh[k];
            acc0 += hs[0][k] * w; acc1 += hs[1][k] * w;
            acc2 += hs[2][k] * w; acc3 += hs[3][k] * w;
        }
        gs[0][tid] = acc0; gs[1][tid] = acc1; gs[2][tid] = acc2; gs[3][tid] = acc3;
        __syncthreads();

        // gate order i,f,g,o; tid -> (batch, cell)
        {
            int b = tid >> 8, cell = tid & 255;
            float ig = sigmoidf_(gs[b][cell]);
            float fg = sigmoidf_(gs[b][RH_ + cell]);
            float gg = tanhf(gs[b][2 * RH_ + cell]);
            float og = sigmoidf_(gs[b][3 * RH_ + cell]);
            float c  = fg * cs[b][cell] + ig * gg;
            cs[b][cell] = c;
            hs[b][cell] = og * tanhf(c);
        }
        __syncthreads();
    }

    int b = tid >> 8, cell = tid & 255;
    feat[b * (2 * RH_) + dir * RH_ + cell] = hs[b][cell];
}

// ---------------- VAE heads + Poincare projection ----------------
__global__ void head_kernel(const float* __restrict__ feat,
                            const float* __restrict__ muW, const float* __restrict__ mub,
                            const float* __restrict__ lvW, const float* __restrict__ lvb,
                            float* __restrict__ out) {
    __shared__ float red[LD_];
    const int b = blockIdx.x, d = threadIdx.x;    // 64 threads
    const float* f  = feat + b * (2 * RH_);
    const float* mw = muW + d * (2 * RH_);
    const float* lw = lvW + d * (2 * RH_);
    float m = mub[d], l = lvb[d];
    for (int k = 0; k < 2 * RH_; ++k) { m += f[k] * mw[k]; l += f[k] * lw[k]; }
    out[BS_ * LD_ + b * LD_ + d] = l;             // logvar (second output)

    red[d] = m * m;
    __syncthreads();
    for (int s = 32; s > 0; s >>= 1) {
        if (d < s) red[d] += red[d + s];
        __syncthreads();
    }
    const float maxnorm = 1.0f - 4e-3f;
    float nrm = sqrtf(red[0]);
    float scale = (nrm > maxnorm) ? (maxnorm / nrm) : 1.0f;
    out[b * LD_ + d] = m * scale;                 // mu (first output)
}

// ---------------- host-side layer driver ----------------
static void run_gat_layer(const float* x, const float* W, const float* asrc,
                          const float* adst, const float* bias, const int* ei,
                          float* hbuf, float* aggbuf,
                          float* al_s, float* al_d, unsigned* mkey, float* den,
                          int K, int H, hipStream_t stream) {
    const int Nc = H * HID_;
    wmma_matmul<<<dim3(Nc / 16, NNODE_ / GEMM_ROWS), 32 * GEMM_WAVES, 0, stream>>>(
        x, W, hbuf, K, Nc);

    const int nh = NNODE_ * H;
    attn_logits<<<(nh + 255) / 256, 256, 0, stream>>>(hbuf, asrc, adst, al_s, al_d, H);
    fill_u32<<<(nh + 255) / 256, 256, 0, stream>>>(mkey, 0u, nh);
    fill_f32<<<(nh + 255) / 256, 256, 0, stream>>>(den, 0.f, nh);

    const int nagg = NNODE_ * Nc;
    fill_f32<<<(nagg + 255) / 256, 256, 0, stream>>>(aggbuf, 0.f, nagg);

    edge_max<<<(ETOT_ + 255) / 256, 256, 0, stream>>>(ei, al_s, al_d, mkey, H);
    edge_den<<<(ETOT_ + 255) / 256, 256, 0, stream>>>(ei, al_s, al_d, mkey, den, H);
    const int eh = ETOT_ * H;
    edge_agg<<<(eh + 255) / 256, 256, 0, stream>>>(ei, hbuf, al_s, al_d, mkey, den, aggbuf, H);

    bias_relu<<<(nagg + 255) / 256, 256, 0, stream>>>(aggbuf, bias, Nc, nagg);
}

extern "C" void kernel_launch(void* const* d_in, const int* in_sizes, int n_in,
                              void* d_out, int out_size, void* d_ws, size_t ws_size,
                              hipStream_t stream) {
    const float* x     = (const float*)d_in[0];
    const int*   ei    = (const int*)d_in[1];
    const float* g0W   = (const float*)d_in[2];
    const float* g0as  = (const float*)d_in[3];
    const float* g0ad  = (const float*)d_in[4];
    const float* g0b   = (const float*)d_in[5];
    const float* g1W   = (const float*)d_in[6];
    const float* g1as  = (const float*)d_in[7];
    const float* g1ad  = (const float*)d_in[8];
    const float* g1b   = (const float*)d_in[9];
    const float* g2W   = (const float*)d_in[10];
    const float* g2as  = (const float*)d_in[11];
    const float* g2ad  = (const float*)d_in[12];
    const float* g2b   = (const float*)d_in[13];
    const float* WihF  = (const float*)d_in[14];
    const float* WhhF  = (const float*)d_in[15];
    const float* bihF  = (const float*)d_in[16];
    const float* bhhF  = (const float*)d_in[17];
    const float* WihB  = (const float*)d_in[18];
    const float* WhhB  = (const float*)d_in[19];
    const float* bihB  = (const float*)d_in[20];
    const float* bhhB  = (const float*)d_in[21];
    const float* muW   = (const float*)d_in[22];
    const float* mub   = (const float*)d_in[23];
    const float* lvW   = (const float*)d_in[24];
    const float* lvb   = (const float*)d_in[25];

    // ---- workspace carving ----
    float*    bufA = (float*)d_ws;                          // [N,256] h (pre-attn)
    float*    bufB = bufA + (size_t)NNODE_ * 256;           // [N,256] agg / layer io
    float*    al_s = bufB + (size_t)NNODE_ * 256;           // [N,H]
    float*    al_d = al_s + (size_t)NNODE_ * HEADS_;
    unsigned* mkey = (unsigned*)(al_d + (size_t)NNODE_ * HEADS_);
    float*    den  = (float*)(mkey + (size_t)NNODE_ * HEADS_);
    float*    embs = den + (size_t)NNODE_ * HEADS_;         // [NG,HID]
    float*    feat = embs + (size_t)NG_ * HID_;             // [BS,2*RH]

    // GAT layer 0: [N,16] -> [N,256]
    run_gat_layer(x,    g0W, g0as, g0ad, g0b, ei, bufA, bufB,
                  al_s, al_d, mkey, den, NF_, HEADS_, stream);
    // GAT layer 1: [N,256] -> [N,256]
    run_gat_layer(bufB, g1W, g1as, g1ad, g1b, ei, bufA, bufB,
                  al_s, al_d, mkey, den, HEADS_ * HID_, HEADS_, stream);
    // GAT layer 2: [N,256] -> [N,64], single head
    run_gat_layer(bufB, g2W, g2as, g2ad, g2b, ei, bufA, bufB,
                  al_s, al_d, mkey, den, HEADS_ * HID_, 1, stream);

    // graph pooling
    pool_kernel<<<NG_, HID_, 0, stream>>>(bufB, embs);

    // bidirectional LSTM (block 0 = fwd, block 1 = bwd)
    lstm_kernel<<<2, 1024, 0, stream>>>(embs, WihF, WhhF, bihF, bhhF,
                                        WihB, WhhB, bihB, bhhB, feat);

    // heads + Poincare projection -> d_out = [mu(4x64) | logvar(4x64)]
    head_kernel<<<BS_, LD_, 0, stream>>>(feat, muW, mub, lvW, lvb, (float*)d_out);
}